// KeywordsPreservingGenerator_69801808495241
// MI455X (gfx1250) — compile-verified
//
#include <hip/hip_runtime.h>
#include <hip/hip_bf16.h>

// ---------------------------------------------------------------------------
// Fused KeywordsPreservingGenerator for gfx1250 (MI455X, wave32, WMMA).
// f32 math throughout (matches reference) via V_WMMA_F32_16X16X4_F32.
// Gumbel tile staged to LDS with CDNA5 async Global->LDS DMA (ASYNCcnt).
// ---------------------------------------------------------------------------

typedef float v2f __attribute__((ext_vector_type(2)));
typedef float v8f __attribute__((ext_vector_type(8)));

#define NROW_TOT 16384   // BS*LS
#define DD       128
#define HH       512
#define NTGT     2048
#define MC       20
#define ROWS     16      // rows per workgroup
#define NTHR     256     // 8 waves of 32
#define NBLK     (NROW_TOT / ROWS)

// output layout (floats, concatenated in reference return order)
#define OUT_MASK 0
#define OUT_LOGP 16384
#define OUT_WORD 32768
#define OUT_CHAR 49152
#define OUT_EMB  376832
#define OUT_LOSS 2473984

// LDS layout in floats (padded strides: stride % 64 == 4 -> conflict-free
// column reads across the 16 rows)
#define SFT_LD 2052
#define HDN_LD 516
#define XS_LD  132
#define GUM_LD 2052
#define OFF_SFT 0
#define OFF_HDN (OFF_SFT + ROWS * SFT_LD)          // 32832
#define OFF_XS  (OFF_HDN + ROWS * HDN_LD)          // 41088
#define OFF_TGT (OFF_XS + ROWS * XS_LD)            // 43200 (ints)
#define OFF_RED (OFF_TGT + NTGT)                   // 45248
#define OFF_REDB (OFF_RED + 256)                   // 45504
#define OFF_REDC (OFF_REDB + 256)                  // 45760 (float or int)
#define OFF_STATS (OFF_REDC + 256)                 // 46016
#define OFF_GUM (OFF_STATS + 128)                  // 46144
#define LDS_FLOATS (OFF_GUM + ROWS * GUM_LD)       // 78976
#define LDS_BYTES (LDS_FLOATS * 4)                 // 315904 B < 320KB/WGP

__device__ __forceinline__ v8f wmma_f32(v2f a, v2f b, v8f c) {
  // V_WMMA_F32_16X16X4_F32 : D = A(16x4) * B(4x16) + C(16x16), wave32
  return __builtin_amdgcn_wmma_f32_16x16x4_f32(
      /*neg_a=*/false, a, /*neg_b=*/false, b,
      /*c_mod=*/(short)0, c, /*reuse_a=*/false, /*reuse_b=*/false);
}

// Async Global -> LDS copy, 16B per lane, tracked with ASYNCcnt.
// VGLOBAL encoding: vdst = LDS byte offset (within this WG's allocation),
// vaddr = 64-bit global address, saddr = off.
__device__ __forceinline__ void async_g2l_b128(unsigned lds_byte,
                                               const float* gaddr) {
  asm volatile("global_load_async_to_lds_b128 %0, %1, off"
               :: "v"(lds_byte), "v"(gaddr) : "memory");
}

__device__ __forceinline__ void wait_asynccnt0() {
#if defined(__has_builtin)
#if __has_builtin(__builtin_amdgcn_s_wait_asynccnt)
  __builtin_amdgcn_s_wait_asynccnt(0);
#else
  asm volatile("s_wait_asynccnt 0x0" ::: "memory");
#endif
#else
  asm volatile("s_wait_asynccnt 0x0" ::: "memory");
#endif
}

__global__ __launch_bounds__(NTHR)
void kpg_fused_kernel(const int* __restrict__ inp_word,
                      const unsigned char* __restrict__ keyword_table,
                      const int* __restrict__ tgt_ids,
                      const int* __restrict__ lut,
                      const float* __restrict__ gum,
                      const float* __restrict__ W,
                      const float* __restrict__ W1,
                      const float* __restrict__ b1,
                      const float* __restrict__ W2,
                      const float* __restrict__ b2,
                      float* __restrict__ out,
                      float* __restrict__ ws) {
  extern __shared__ float sh[];
  float* sSft = sh + OFF_SFT;     // [16][2052] logits -> exp((l+g)/T - m2)
  float* sHdn = sh + OFF_HDN;     // [16][516]
  float* sXs  = sh + OFF_XS;      // [16][132]
  int*   sTgt = (int*)(sh + OFF_TGT);
  float* sRed = sh + OFF_RED;     // [16][16] partials
  float* sRedB = sh + OFF_REDB;
  float* sRedC = sh + OFF_REDC;
  int*   sRedI = (int*)(sh + OFF_REDC);
  float* sGum = sh + OFF_GUM;     // [16][2052] async-staged gumbel tile
  float* stM1   = sh + OFF_STATS;        // [16]
  float* stM2   = stM1 + 16;
  float* stRden = stM1 + 32;
  float* stEnt  = stM1 + 48;
  float* stMask = stM1 + 64;
  int*   stIdx  = (int*)(stM1 + 80);     // [0..15]=argmax col
  int*   stWrd0 = stIdx + 16;            // original word ids
  int*   stWrdF = stIdx + 32;            // final word ids

  const int tid  = threadIdx.x;
  const int wave = tid >> 5;
  const int lane = tid & 31;
  const int ln   = lane & 15;
  const int hi   = lane >> 4;            // half-wave select (K offset +2)
  const int rowBase = blockIdx.x * ROWS;

  // ---- kick off async gumbel tile copy (overlaps GEMM1 + GEMM2) -----------
  {
    const float* gbase = gum + (size_t)rowBase * NTGT;
    for (int i = tid * 4; i < ROWS * NTGT; i += NTHR * 4) {
      const int r = i >> 11;             // i / NTGT
      const int c = i & (NTGT - 1);
      const unsigned dstB = (unsigned)((OFF_GUM + r * GUM_LD + c) * 4);
      async_g2l_b128(dstB, gbase + i);   // 16B per lane, fully coalesced
    }
  }

  // ---- load phase: tgt_ids, embedding tile, per-row word/mask -------------
  for (int i = tid; i < NTGT; i += NTHR) sTgt[i] = tgt_ids[i];
  for (int i = tid; i < ROWS * DD; i += NTHR) {
    int r = i >> 7, c = i & 127;
    int w = inp_word[rowBase + r];
    sXs[r * XS_LD + c] = W[(size_t)w * DD + c];
  }
  if (tid < ROWS) {
    int w = inp_word[rowBase + tid];
    stWrd0[tid] = w;
    stMask[tid] = keyword_table[w] ? 1.0f : 0.0f;
  }
  __syncthreads();

  // ---- GEMM1: hdn = relu(X[16,128] @ W1[128,512] + b1) --------------------
  for (int i = 0; i < 4; ++i) {          // 32 N-tiles, 4 per wave
    const int n0 = (wave * 4 + i) * 16;
    v8f acc = {};
#pragma unroll 4
    for (int k = 0; k < DD; k += 4) {
      const int kk = k + 2 * hi;
      v2f a, b;
      a[0] = sXs[ln * XS_LD + kk];
      a[1] = sXs[ln * XS_LD + kk + 1];
      b[0] = W1[(size_t)kk * HH + n0 + ln];
      b[1] = W1[(size_t)(kk + 1) * HH + n0 + ln];
      acc = wmma_f32(a, b, acc);
    }
    const float bias = b1[n0 + ln];
#pragma unroll
    for (int j = 0; j < 8; ++j) {
      const int m = j + 8 * hi;
      sHdn[m * HDN_LD + n0 + ln] = fmaxf(acc[j] + bias, 0.0f);
    }
  }
  __syncthreads();

  // ---- GEMM2: logits = hdn[16,512] @ W2[512,2048] + b2 --------------------
  for (int i = 0; i < 16; ++i) {         // 128 N-tiles, 16 per wave
    const int n0 = (wave * 16 + i) * 16;
    __builtin_prefetch(&W2[(size_t)(2 * hi) * NTGT + n0 + ln], 0, 1);
    v8f acc = {};
#pragma unroll 4
    for (int k = 0; k < HH; k += 4) {
      const int kk = k + 2 * hi;
      v2f a, b;
      a[0] = sHdn[ln * HDN_LD + kk];
      a[1] = sHdn[ln * HDN_LD + kk + 1];
      b[0] = W2[(size_t)kk * NTGT + n0 + ln];
      b[1] = W2[(size_t)(kk + 1) * NTGT + n0 + ln];
      acc = wmma_f32(a, b, acc);
    }
    const float bias = b2[n0 + ln];
#pragma unroll
    for (int j = 0; j < 8; ++j) {
      const int m = j + 8 * hi;
      sSft[m * SFT_LD + n0 + ln] = acc[j] + bias;
    }
  }
  wait_asynccnt0();                      // gumbel tile landed in LDS
  __syncthreads();

  // ---- pass 1: row max (log_p) and gumbel-argmax/max ----------------------
  {
    const int r = tid >> 4, t = tid & 15;
    float m1 = -3.0e38f, m2 = -3.0e38f;
    int ai = 0;
    for (int c = t; c < NTGT; c += 16) {
      const float l = sSft[r * SFT_LD + c];
      const float z = (l + sGum[r * GUM_LD + c]) * 2.0f;   // /TEMP, TEMP=0.5
      m1 = fmaxf(m1, l);
      if (z > m2) { m2 = z; ai = c; }
    }
    sRed[r * 16 + t] = m1;
    sRedB[r * 16 + t] = m2;
    sRedI[r * 16 + t] = ai;
  }
  __syncthreads();
  if (tid < ROWS) {
    const int r = tid;
    float m1 = -3.0e38f, m2 = -3.0e38f; int ai = 0;
    for (int t = 0; t < 16; ++t) {
      m1 = fmaxf(m1, sRed[r * 16 + t]);
      const float z = sRedB[r * 16 + t];
      if (z > m2) { m2 = z; ai = sRedI[r * 16 + t]; }
    }
    stM1[r] = m1; stM2[r] = m2; stIdx[r] = ai;
  }
  __syncthreads();

  // ---- pass 2: softmax sums, entropy pieces; store e2 in-place ------------
  {
    const int r = tid >> 4, t = tid & 15;
    const float m1 = stM1[r], m2 = stM2[r];
    float s1 = 0.0f, s2 = 0.0f, s3 = 0.0f;
    for (int c = t; c < NTGT; c += 16) {
      const float l = sSft[r * SFT_LD + c];
      const float e1 = __expf(l - m1);
      s1 += e1;
      s2 += l * e1;
      const float e2 = __expf((l + sGum[r * GUM_LD + c]) * 2.0f - m2);
      s3 += e2;
      sSft[r * SFT_LD + c] = e2;                     // reuse as gumbel numer
    }
    sRed[r * 16 + t] = s1;
    sRedB[r * 16 + t] = s2;
    sRedC[r * 16 + t] = s3;
  }
  __syncthreads();
  if (tid < ROWS) {
    const int r = tid;
    float s1 = 0.0f, s2 = 0.0f, s3 = 0.0f;
    for (int t = 0; t < 16; ++t) {
      s1 += sRed[r * 16 + t]; s2 += sRedB[r * 16 + t]; s3 += sRedC[r * 16 + t];
    }
    const float ent = (stM1[r] + __logf(s1)) - s2 / s1;
    stEnt[r] = ent;
    stRden[r] = 1.0f / s3;
    const float mk = stMask[r];
    const int wF = (mk > 0.0f) ? sTgt[stIdx[r]] : stWrd0[r];
    stWrdF[r] = wF;
    const int R = rowBase + r;
    out[OUT_MASK + R] = mk;
    out[OUT_LOGP + R] = (mk > 0.0f) ? stM1[r] : 0.0f;
    out[OUT_WORD + R] = (float)wF;
  }
  __syncthreads();

  // ---- char gather + block contribution to hloss --------------------------
  for (int i = tid; i < ROWS * MC; i += NTHR) {
    const int r = i / MC, j = i - r * MC;
    out[OUT_CHAR + (size_t)(rowBase + r) * MC + j] =
        (float)lut[(size_t)stWrdF[r] * MC + j];
  }
  if (tid == 0) {
    float es = 0.0f, cnt = 0.0f;
    for (int r = 0; r < ROWS; ++r) { es += stEnt[r] * stMask[r]; cnt += stMask[r]; }
    atomicAdd(&ws[0], es);
    atomicAdd(&ws[1], cnt);
  }

  // ---- GEMM3: x_emb = sft[16,2048] @ W[tgt_ids][2048,128], fold 1/den -----
  {
    const int n0 = wave * 16;            // 8 N-tiles of 16 cover D=128
    v8f acc = {};
#pragma unroll 4
    for (int k = 0; k < NTGT; k += 4) {
      const int kk = k + 2 * hi;
      v2f a, b;
      a[0] = sSft[ln * SFT_LD + kk];
      a[1] = sSft[ln * SFT_LD + kk + 1];
      const int t0 = sTgt[kk];
      const int t1 = sTgt[kk + 1];
      b[0] = W[(size_t)t0 * DD + n0 + ln];
      b[1] = W[(size_t)t1 * DD + n0 + ln];
      acc = wmma_f32(a, b, acc);
    }
#pragma unroll
    for (int j = 0; j < 8; ++j) {
      const int m = j + 8 * hi;
      const int col = n0 + ln;
      const float emb = (stMask[m] > 0.0f) ? acc[j] * stRden[m]
                                           : sXs[m * XS_LD + col];
      out[OUT_EMB + (size_t)(rowBase + m) * DD + col] = emb;
    }
  }
}

__global__ void kpg_init_kernel(float* ws) {
  if (threadIdx.x < 2) ws[threadIdx.x] = 0.0f;
}

__global__ void kpg_final_kernel(const float* __restrict__ ws,
                                 float* __restrict__ out) {
  if (threadIdx.x == 0) {
    out[OUT_LOSS] = ws[0] / fmaxf(ws[1], 1.0f) * 0.03f;
  }
}

extern "C" void kernel_launch(void* const* d_in, const int* in_sizes, int n_in,
                              void* d_out, int out_size, void* d_ws, size_t ws_size,
                              hipStream_t stream) {
  // setup_inputs order:
  // 0 inp_word [32,512] i32        1 inp_char (unused)   2 inp_pos (unused)
  // 3 keyword_table [100000] bool  4 tgt_ids [2048] i32  5 lut [100000,20] i32
  // 6 gumbel [32,512,2048] f32     7 W [100000,128] f32  8 W1 [128,512] f32
  // 9 b1 [512] f32                10 W2 [512,2048] f32  11 b2 [2048] f32
  const int*           inp_word = (const int*)d_in[0];
  const unsigned char* kwtab    = (const unsigned char*)d_in[3];  // jax bool
  const int*           tgt      = (const int*)d_in[4];
  const int*           lut      = (const int*)d_in[5];
  const float*         gum      = (const float*)d_in[6];
  const float*         W        = (const float*)d_in[7];
  const float*         W1       = (const float*)d_in[8];
  const float*         b1       = (const float*)d_in[9];
  const float*         W2       = (const float*)d_in[10];
  const float*         b2       = (const float*)d_in[11];
  float* out = (float*)d_out;
  float* ws  = (float*)d_ws;

  kpg_init_kernel<<<1, 32, 0, stream>>>(ws);
  kpg_fused_kernel<<<NBLK, NTHR, LDS_BYTES, stream>>>(
      inp_word, kwtab, tgt, lut, gum, W, W1, b1, W2, b2, out, ws);
  kpg_final_kernel<<<1, 32, 0, stream>>>(ws, out);
}